// SeqFeatLayer_49237505081482
// MI455X (gfx1250) — compile-verified
//
#include <hip/hip_runtime.h>
#include <stdint.h>

// Problem constants (match reference; actual sizes taken from in_sizes)
#define EPS_F 1e-6f
#define CHUNK 2048          // edges staged per block
#define BLK   256           // 8 waves of 32 on CDNA5
#define EPT   (CHUNK / BLK) // 8 edges per thread

// ---------------------------------------------------------------------------
// Kernel 1: zero the output buffer (it is reused as the atomic accumulator).
// d_out layout per node (6 floats): [sum_dx, sum_dy, sum_ndx, sum_ndy, cnt, _]
// ---------------------------------------------------------------------------
__global__ __launch_bounds__(256) void zero_out_kernel(float* __restrict__ out,
                                                       int n) {
  int i = blockIdx.x * blockDim.x + threadIdx.x;
  if (i < n) out[i] = 0.0f;
}

// ---------------------------------------------------------------------------
// Kernel 2: edge scatter. Stage the two edge-index rows for this block's
// chunk into LDS with CDNA5 async global->LDS loads, then gather pos and
// atomically accumulate messages into d_out[dst*6 + k].
// ---------------------------------------------------------------------------
__global__ __launch_bounds__(BLK) void edge_scatter_kernel(
    const float2* __restrict__ pos,
    const int*    __restrict__ src_idx,   // edge_index row 0
    const int*    __restrict__ dst_idx,   // edge_index row 1
    float*        __restrict__ acc,       // = d_out, stride 6 per node
    int n_edges) {
  __shared__ int sbuf[CHUNK];
  __shared__ int dbuf[CHUNK];

  const int  t           = threadIdx.x;
  const long chunk_start = (long)blockIdx.x * CHUNK;
  long rem_l = (long)n_edges - chunk_start;
  const int rem = (rem_l > CHUNK) ? CHUNK : (int)rem_l;   // > 0 by grid sizing

  // --- Stage indices into LDS: 16B (4 ints) per lane per instruction. ---
  // LDS byte offsets: low 32 bits of the flat address of a __shared__ object
  // are the LDS byte address (aperture truncation rule).
  const uint32_t s_lds0 = (uint32_t)(uintptr_t)sbuf;
  const uint32_t d_lds0 = (uint32_t)(uintptr_t)dbuf;

#pragma unroll
  for (int p = 0; p < 2; ++p) {
    const int idx = p * (BLK * 4) + t * 4;  // 4 consecutive ints per thread
    if (idx + 4 <= rem) {
      const uint32_t goff  = (uint32_t)((chunk_start + (long)idx) * 4);
      const uint32_t lds_s = s_lds0 + (uint32_t)idx * 4u;
      const uint32_t lds_d = d_lds0 + (uint32_t)idx * 4u;
      // GVS mode: reads from the 64-bit SADDR base plus the byte offset in VADDR, writes to the LDS address in vdst
      asm volatile("global_load_async_to_lds_b128 %0, %1, %2"
                   :: "v"(lds_s), "v"(goff), "s"(src_idx) : "memory");
      asm volatile("global_load_async_to_lds_b128 %0, %1, %2"
                   :: "v"(lds_d), "v"(goff), "s"(dst_idx) : "memory");
    } else if (idx < rem) {
      // partial 16B granule at the tail: plain loads + LDS stores
      for (int k = idx; k < rem && k < idx + 4; ++k) {
        sbuf[k] = src_idx[chunk_start + k];
        dbuf[k] = dst_idx[chunk_start + k];
      }
    }
  }

  // Wait for this wave's async transfers, then block barrier so every wave
  // sees every wave's staged data.
  asm volatile("s_wait_asynccnt 0x0" ::: "memory");
  __syncthreads();

  // --- Process 8 edges per thread; lane-consecutive LDS reads (no conflicts).
#pragma unroll
  for (int i = 0; i < EPT; ++i) {
    const int e = i * BLK + t;
    if (e < rem) {
      const int s = sbuf[e];
      const int d = dbuf[e];
      const float2 ps = pos[s];            // random 8B gathers; pos (4MB) is
      const float2 pd = pos[d];            // L2-resident on MI455X (192MB L2)
      const float dx = ps.x - pd.x;
      const float dy = ps.y - pd.y;
      const float r  = sqrtf(fmaf(dx, dx, dy * dy));
      const float inv = 1.0f / (r + EPS_F);

      float* a = acc + (long)d * 6;        // 5 atomics within 20 bytes -> good
      unsafeAtomicAdd(a + 0, dx);          // L2 sector locality
      unsafeAtomicAdd(a + 1, dy);
      unsafeAtomicAdd(a + 2, dx * inv);
      unsafeAtomicAdd(a + 3, dy * inv);
      unsafeAtomicAdd(a + 4, 1.0f);
    }
  }
}

// ---------------------------------------------------------------------------
// Kernel 3: per-node mean + concat pos.
// ---------------------------------------------------------------------------
__global__ __launch_bounds__(256) void finalize_kernel(
    const float2* __restrict__ pos, float* __restrict__ out, int n_nodes) {
  const int i = blockIdx.x * blockDim.x + threadIdx.x;
  if (i >= n_nodes) return;
  float* o = out + (long)i * 6;
  const float c   = o[4];
  const float inv = 1.0f / fmaxf(c, 1.0f);
  const float s0 = o[0], s1 = o[1], s2 = o[2], s3 = o[3];
  const float2 p = pos[i];
  o[0] = s0 * inv;
  o[1] = s1 * inv;
  o[2] = s2 * inv;
  o[3] = s3 * inv;
  o[4] = p.x;
  o[5] = p.y;
}

// ---------------------------------------------------------------------------
extern "C" void kernel_launch(void* const* d_in, const int* in_sizes, int n_in,
                              void* d_out, int out_size, void* d_ws,
                              size_t ws_size, hipStream_t stream) {
  const float* pos_f = (const float*)d_in[0];  // (N_NODES, 2) float32
  const int*   ei    = (const int*)d_in[1];    // (2, N_EDGES) int
  const int n_nodes = in_sizes[0] / 2;
  const int n_edges = in_sizes[1] / 2;
  float* out = (float*)d_out;                  // (N_NODES, 6) float32

  // 1) zero accumulators (d_out itself); required every call (graph replay).
  const int ztot = n_nodes * 6;
  zero_out_kernel<<<(ztot + 255) / 256, 256, 0, stream>>>(out, ztot);

  // 2) scatter-accumulate over all edges.
  const int nblocks = (n_edges + CHUNK - 1) / CHUNK;
  edge_scatter_kernel<<<nblocks, BLK, 0, stream>>>(
      (const float2*)pos_f, ei, ei + n_edges, out, n_edges);

  // 3) normalize + append pos.
  finalize_kernel<<<(n_nodes + 255) / 256, 256, 0, stream>>>(
      (const float2*)pos_f, out, n_nodes);
}